// SparseRuleLayer_16140487099033
// MI455X (gfx1250) — compile-verified
//
#include <hip/hip_runtime.h>
#include <hip/hip_bf16.h>
#include <math.h>

// ---------------------------------------------------------------------------
// Problem constants (from reference setup_inputs)
// ---------------------------------------------------------------------------
#define NN   256          // number of objects
#define DD   512          // attr dim
#define KK   8            // K_OPS
#define NP   4            // N_PARAMS
#define JC   48           // padded combined output cols (8 logits + 32 params + 8 pad)
#define HH   1024
#define WW   1024
#define HWPX (HH*WW)

typedef float v2f __attribute__((ext_vector_type(2)));
typedef float v8f __attribute__((ext_vector_type(8)));

// ---------------------------------------------------------------------------
// Kernel 0: build combined padded weight matrix Wcomb (DD x JC)
//   col j<8   -> selector_w[d][j]; col 8..39 -> param_w[d][j-8]; col>=40 -> 0
// ---------------------------------------------------------------------------
__global__ void build_wcomb_kernel(const float* __restrict__ sw,
                                   const float* __restrict__ pw,
                                   float* __restrict__ wcomb) {
    int idx = blockIdx.x * blockDim.x + threadIdx.x;
    if (idx >= DD * JC) return;
    int d = idx / JC, j = idx % JC;
    float v = 0.0f;
    if (j < KK)              v = sw[d * KK + j];
    else if (j < KK + KK*NP) v = pw[d * (KK*NP) + (j - KK)];
    wcomb[idx] = v;
}

// ---------------------------------------------------------------------------
// Kernel 1: C(256x48) = attr(256x512) @ Wcomb(512x48) via V_WMMA_F32_16X16X4_F32
// One wave per 16x16 tile: 16 row-tiles * 3 col-tiles = 48 blocks of 32 threads.
// ISA lane layouts (7.12.2):
//   A 16x4 f32 : lanes 0-15 -> M=lane, regs {K=0,K=1}; lanes 16-31 -> {K=2,K=3}
//   B 4x16 f32 : lanes 0-15 -> N=lane, regs {K=0,K=1}; lanes 16-31 -> {K=2,K=3}
//   C/D 16x16  : reg r: lanes 0-15 -> M=r, lanes 16-31 -> M=8+r; N=lane%16
// ---------------------------------------------------------------------------
__global__ __launch_bounds__(32)
void wmma_gemm_kernel(const float* __restrict__ attr,
                      const float* __restrict__ wcomb,
                      float* __restrict__ cout) {
    const int tile = blockIdx.x;        // 0..47
    const int tm   = tile / 3;
    const int tn   = tile % 3;
    const int lane = threadIdx.x;       // 0..31
    const int half = lane >> 4;         // 0: K pair {0,1}; 1: K pair {2,3}
    const int l16  = lane & 15;

    const int arow = tm * 16 + l16;
    const int bcol = tn * 16 + l16;

    v8f acc = {};
    for (int k0 = 0; k0 < DD; k0 += 4) {
        const float* ap = attr  + (size_t)arow * DD + (k0 + 2 * half);
        v2f a = { ap[0], ap[1] };
        const float* bp = wcomb + (size_t)(k0 + 2 * half) * JC + bcol;
        v2f b = { bp[0], bp[JC] };
        acc = __builtin_amdgcn_wmma_f32_16x16x4_f32(
                  false, a, false, b, (short)0, acc, false, false);
    }
#pragma unroll
    for (int r = 0; r < 8; ++r) {
        int m = r + 8 * half;
        cout[(size_t)(tm * 16 + m) * JC + bcol] = acc[r];
    }
}

// ---------------------------------------------------------------------------
// Kernel 2: per object n: argmax(logits + gumbel) -> k_sel, gather params,
// precompute op class + affine coefficients (cv' = a*cv + b for 7 of 8 ops).
// ---------------------------------------------------------------------------
__global__ void select_kernel(const float* __restrict__ cout,
                              const float* __restrict__ selb,
                              const float* __restrict__ gumbel,
                              const float* __restrict__ parb,
                              int*   __restrict__ flags,
                              float* __restrict__ coef) {
    int n = blockIdx.x * blockDim.x + threadIdx.x;
    if (n >= NN) return;
    const float* row = cout + (size_t)n * JC;

    float best = -INFINITY;
    int ksel = 0;
#pragma unroll
    for (int k = 0; k < KK; ++k) {
        float v = row[k] + selb[k] + gumbel[n * KK + k];
        if (v > best) { best = v; ksel = k; }   // first-max-wins (jnp.argmax)
    }

    float p0 = row[KK + ksel * NP + 0] + parb[ksel * NP + 0];
    float p1 = row[KK + ksel * NP + 1] + parb[ksel * NP + 1];
    float p2 = row[KK + ksel * NP + 2] + parb[ksel * NP + 2];
    float p3 = row[KK + ksel * NP + 3] + parb[ksel * NP + 3];

    int   f = 0;
    float a = 1.0f, b = 0.0f, c = 0.0f, d = 0.0f;
    switch (ksel) {
        case 0: a = 0.0f;  b = p0;          break;  // recolor
        case 1: a = 1.0f;  b = p0;          break;  // add
        case 2: a = p1;    b = 0.0f;        break;  // scale (uses p[1]!)
        case 3: a = p0;    b = p1;          break;  // affine
        case 4: {                                    // blend
            float s = 1.0f / (1.0f + __expf(-p2));
            a = 1.0f - s; b = s * p0;       break;
        }
        case 5: a = -1.0f; b = p0;          break;  // invert
        case 6: f = 1; a = p0; b = p1; c = p2; d = p3; break; // tanh_mod
        case 7: a = 1.0f;  b = p0 * p1;     break;  // offset
    }
    flags[n] = f;
    coef[4 * n + 0] = a;
    coef[4 * n + 1] = b;
    coef[4 * n + 2] = c;
    coef[4 * n + 3] = d;
}

// ---------------------------------------------------------------------------
// Kernel 3: streaming canvas pass (dominant: 268 MB mask stream).
// 8 px/thread; one 8-byte NT mask load always in flight (prefetch address
// wraps with (n+1)&255 -> single loop body, no epilogue). Exact select per
// pixel: bfe + v_cmp + v_fma + v_cndmask (4 VALU/px, VOPD-pairable).
// Per-n op branch is wave-uniform (coefficients broadcast from LDS).
// ---------------------------------------------------------------------------
__global__ __launch_bounds__(256)
void canvas_kernel(const float* __restrict__ canvas_in,
                   const unsigned char* __restrict__ masks,
                   const int*   __restrict__ flags,
                   const float* __restrict__ coef,
                   float* __restrict__ out) {
    __shared__ int   sflag[NN];
    __shared__ float scoef[NN * 4];
    for (int i = threadIdx.x; i < NN;     i += 256) sflag[i] = flags[i];
    for (int i = threadIdx.x; i < NN * 4; i += 256) scoef[i] = coef[i];
    __syncthreads();

    const int base = (blockIdx.x * 256 + threadIdx.x) * 8;   // 8 px/thread

    float cv[8];
    {
        const float4 c0 = *(const float4*)(canvas_in + base);
        const float4 c1 = *(const float4*)(canvas_in + base + 4);
        cv[0]=c0.x; cv[1]=c0.y; cv[2]=c0.z; cv[3]=c0.w;
        cv[4]=c1.x; cv[5]=c1.y; cv[6]=c1.z; cv[7]=c1.w;
    }

    const unsigned char* mp = masks + base;

    unsigned long long mw = __builtin_nontemporal_load((const unsigned long long*)mp);
    for (int n = 0; n < NN; ++n) {
        const unsigned long long cur = mw;
        // prefetch next row; wraps to row 0 on the last iteration (harmless)
        mw = __builtin_nontemporal_load(
                 (const unsigned long long*)(mp + (size_t)((n + 1) & (NN - 1)) * HWPX));

        const unsigned int w0 = (unsigned int)cur;          // pixels 0..3
        const unsigned int w1 = (unsigned int)(cur >> 32);  // pixels 4..7

        if (sflag[n] == 0) {                                // wave-uniform
            const float a = scoef[4*n + 0];
            const float b = scoef[4*n + 1];
#pragma unroll
            for (int i = 0; i < 8; ++i) {
                const unsigned int w = (i < 4) ? w0 : w1;
                const bool m = ((w >> (8 * (i & 3))) & 0xffu) != 0u;
                const float t = fmaf(a, cv[i], b);
                cv[i] = m ? t : cv[i];
            }
        } else {                                            // tanh_mod (rare)
            const float p0 = scoef[4*n + 0];
            const float p1 = scoef[4*n + 1];
            const float p2 = scoef[4*n + 2];
            const float p3 = scoef[4*n + 3];
#pragma unroll
            for (int i = 0; i < 8; ++i) {
                const unsigned int w = (i < 4) ? w0 : w1;
                const bool m = ((w >> (8 * (i & 3))) & 0xffu) != 0u;
                const float t = fmaf(p2, tanhf(fmaf(p0, cv[i], p1)), p3);
                cv[i] = m ? t : cv[i];
            }
        }
    }

    float4 o0 = { cv[0], cv[1], cv[2], cv[3] };
    float4 o1 = { cv[4], cv[5], cv[6], cv[7] };
    *(float4*)(out + base)     = o0;
    *(float4*)(out + base + 4) = o1;
}

// ---------------------------------------------------------------------------
// Launch
// ---------------------------------------------------------------------------
extern "C" void kernel_launch(void* const* d_in, const int* in_sizes, int n_in,
                              void* d_out, int out_size, void* d_ws, size_t ws_size,
                              hipStream_t stream) {
    const float*         canvas   = (const float*)d_in[0];          // (1024,1024)
    const float*         attr     = (const float*)d_in[1];          // (256,512)
    const unsigned char* masks    = (const unsigned char*)d_in[2];  // (256,1024,1024) bool
    const float*         gumbel   = (const float*)d_in[3];          // (256,8)
    const float*         sel_w    = (const float*)d_in[4];          // (512,8)
    const float*         sel_b    = (const float*)d_in[5];          // (8,)
    const float*         par_w    = (const float*)d_in[6];          // (512,32)
    const float*         par_b    = (const float*)d_in[7];          // (32,)
    float*               out      = (float*)d_out;                  // (1024,1024)

    char* ws = (char*)d_ws;
    float* wcomb = (float*)(ws);                             // 512*48 f32 = 98304 B
    float* cout  = (float*)(ws + 98304);                     // 256*48 f32 = 49152 B
    int*   flags = (int*)  (ws + 98304 + 49152);             // 256 i32   = 1024  B
    float* coef  = (float*)(ws + 98304 + 49152 + 1024);      // 256*4 f32 = 4096  B

    build_wcomb_kernel<<<(DD * JC + 255) / 256, 256, 0, stream>>>(sel_w, par_w, wcomb);
    wmma_gemm_kernel<<<48, 32, 0, stream>>>(attr, wcomb, cout);
    select_kernel<<<1, 256, 0, stream>>>(cout, sel_b, gumbel, par_b, flags, coef);
    canvas_kernel<<<HWPX / (256 * 8), 256, 0, stream>>>(canvas, masks, flags, coef, out);
}